// GAT_FULL_DGL_38225208934552
// MI455X (gfx1250) — compile-verified
//
#include <hip/hip_runtime.h>

typedef __attribute__((ext_vector_type(2))) float v2f;
typedef __attribute__((ext_vector_type(8))) float v8f;

static constexpr int NN   = 100000;
static constexpr int EE   = 1600000;
static constexpr int IND  = 256;
static constexpr int HID  = 128;
static constexpr int HEADS= 8;
static constexpr int DH   = 16;
static constexpr int NCLS = 64;
static constexpr float SLOPE = 0.2f;
static constexpr unsigned NEGKEY = 0x007FFFFFu; // fkey(-inf)

__device__ __forceinline__ unsigned fkey(float f) {
    unsigned u = __float_as_uint(f);
    return (u & 0x80000000u) ? ~u : (u | 0x80000000u);
}
__device__ __forceinline__ float funkey(unsigned k) {
    unsigned u = (k & 0x80000000u) ? (k & 0x7fffffffu) : ~k;
    return __uint_as_float(u);
}
__device__ __forceinline__ float lrelu(float x) { return x > 0.f ? x : SLOPE * x; }
__device__ __forceinline__ float elu1(float x)  { return x > 0.f ? x : expm1f(x); }

// ---------------- init ----------------
__global__ __launch_bounds__(256) void k_initA(unsigned* mkey1, float* s1, float* acc1,
                                               unsigned* mkey2, float* s2) {
    int t = blockIdx.x * 256 + threadIdx.x;
    if (t < NN * HID) acc1[t] = 0.f;
    if (t < NN * HEADS) { mkey1[t] = NEGKEY; s1[t] = 0.f; }
    if (t < NN) { mkey2[t] = NEGKEY; s2[t] = 0.f; }
}
__global__ __launch_bounds__(256) void k_initB(float* acc2) {
    int t = blockIdx.x * 256 + threadIdx.x;
    if (t < NN * NCLS) acc2[t] = 0.f;
}

// ---------------- GEMM1: h1 = X[N,256] @ W1[256,128] (pure WMMA) ----------------
// block = 256 threads = 8 waves; wave w = head/col-tile w; 16 rows per block (N%16==0)
__global__ __launch_bounds__(256) void k_gemm1(const float* __restrict__ X,
                                               const float* __restrict__ W,
                                               float* __restrict__ H) {
    const int wave = threadIdx.x >> 5;
    const int lane = threadIdx.x & 31;
    const int half = lane >> 4;
    const int l16  = lane & 15;
    const int row0 = blockIdx.x * 16;
    const int col  = wave * 16 + l16;

    v8f c = {};
    const float* xrow = X + (size_t)(row0 + l16) * IND; // A-matrix: M = l16
#pragma unroll 4
    for (int kk = 0; kk < IND; kk += 4) {
        const int k = kk + half * 2;
        __builtin_prefetch(xrow + k + IND, 0, 1);        // next row of X (row + 1) (streams from HBM)
        v2f a = *(const v2f*)(xrow + k);                 // K = k, k+1 for this half
        v2f b;
        b.x = W[(size_t)k * HID + col];                  // B: 4x16, rows striped like C/D
        b.y = W[(size_t)(k + 1) * HID + col];
        c = __builtin_amdgcn_wmma_f32_16x16x4_f32(false, a, false, b, (short)0, c, false, false);
    }
#pragma unroll
    for (int v = 0; v < 8; ++v)                          // C/D layout: M = v + 8*half, N = l16
        H[(size_t)(row0 + v + half * 8) * HID + col] = c[v];
}

// ---------------- attention logits, layer 1: el/er[n,h] = sum_d h1[n,h,d]*a[h,d] ----------------
// t = n*8+h, and h1[n*128 + h*16] == h1[t*16]: perfectly coalesced float4 streaming
__global__ __launch_bounds__(256) void k_att1(const float* __restrict__ H,
                                              const float* __restrict__ al,
                                              const float* __restrict__ ar,
                                              float* __restrict__ el, float* __restrict__ er) {
    int t = blockIdx.x * 256 + threadIdx.x;
    if (t >= NN * HEADS) return;
    const int h = t & 7;
    const float4* hp = (const float4*)(H + (size_t)t * DH);
    const float4* ap = (const float4*)(al + h * DH);
    const float4* rp = (const float4*)(ar + h * DH);
    float se = 0.f, sr = 0.f;
#pragma unroll
    for (int i = 0; i < 4; ++i) {
        const float4 hv = hp[i], av = ap[i], rv = rp[i];
        se += hv.x * av.x + hv.y * av.y + hv.z * av.z + hv.w * av.w;
        sr += hv.x * rv.x + hv.y * rv.y + hv.z * rv.z + hv.w * rv.w;
    }
    el[t] = se; er[t] = sr;
}

// ---------------- layer1 edge pass A: segment max ----------------
__global__ __launch_bounds__(256) void k_emax1(const int* __restrict__ src, const int* __restrict__ dst,
                                               const float* __restrict__ el, const float* __restrict__ er,
                                               unsigned* __restrict__ mkey) {
    int t = blockIdx.x * 256 + threadIdx.x;
    if (t >= EE * HEADS) return;
    const int e = t >> 3, h = t & 7;
    const int s = src[e], d = dst[e];
    const float ev = lrelu(el[s * HEADS + h] + er[d * HEADS + h]);
    atomicMax(&mkey[d * HEADS + h], fkey(ev));
}

// ---------------- layer1 edge pass B: p = exp(e - m); s += p; acc += p * h1[src] ----------------
__global__ __launch_bounds__(256) void k_eagg1(const int* __restrict__ src, const int* __restrict__ dst,
                                               const float* __restrict__ el, const float* __restrict__ er,
                                               const unsigned* __restrict__ mkey,
                                               const float* __restrict__ H,
                                               float* __restrict__ ssum, float* __restrict__ acc) {
    int t = blockIdx.x * 256 + threadIdx.x;
    if (t >= EE * HEADS) return;
    const int e = t >> 3, h = t & 7;
    const int s = src[e], d = dst[e];
    const float ev = lrelu(el[s * HEADS + h] + er[d * HEADS + h]);
    float m = funkey(mkey[d * HEADS + h]);
    if (!isfinite(m)) m = 0.f;
    const float p = __expf(ev - m);
    atomicAdd(&ssum[d * HEADS + h], p);
    const float4* hv = (const float4*)(H + (size_t)s * HID + h * DH);
    float* ap = acc + (size_t)d * HID + h * DH;
    float4 a0 = hv[0], a1 = hv[1], a2 = hv[2], a3 = hv[3];
    atomicAdd(ap + 0,  p * a0.x); atomicAdd(ap + 1,  p * a0.y);
    atomicAdd(ap + 2,  p * a0.z); atomicAdd(ap + 3,  p * a0.w);
    atomicAdd(ap + 4,  p * a1.x); atomicAdd(ap + 5,  p * a1.y);
    atomicAdd(ap + 6,  p * a1.z); atomicAdd(ap + 7,  p * a1.w);
    atomicAdd(ap + 8,  p * a2.x); atomicAdd(ap + 9,  p * a2.y);
    atomicAdd(ap + 10, p * a2.z); atomicAdd(ap + 11, p * a2.w);
    atomicAdd(ap + 12, p * a3.x); atomicAdd(ap + 13, p * a3.y);
    atomicAdd(ap + 14, p * a3.z); atomicAdd(ap + 15, p * a3.w);
}

// ---------------- layer1 finalize: h2 = elu(elu(acc/s + b)) ----------------
__global__ __launch_bounds__(256) void k_fin1(const float* __restrict__ acc, const float* __restrict__ ssum,
                                              const float* __restrict__ b, float* __restrict__ h2) {
    int t = blockIdx.x * 256 + threadIdx.x;
    if (t >= NN * HID) return;
    const int n = t >> 7, cidx = t & 127, h = cidx >> 4;
    const float s = ssum[n * HEADS + h];
    float v = (s > 0.f) ? acc[t] / s : 0.f;
    v = elu1(elu1(v + b[cidx]));
    h2[t] = v;
}

// ---------------- GEMM2: h2t = H2[N,128] @ W2[128,64] (pure WMMA) ----------------
// block = 256 = 8 waves = 2 row-tiles x 4 col-tiles; 32 rows per block (N%32==0)
__global__ __launch_bounds__(256) void k_gemm2(const float* __restrict__ H2,
                                               const float* __restrict__ W,
                                               float* __restrict__ HT) {
    const int wave = threadIdx.x >> 5;
    const int lane = threadIdx.x & 31;
    const int half = lane >> 4;
    const int l16  = lane & 15;
    const int rt = wave >> 2, ct = wave & 3;
    const int row0 = blockIdx.x * 32 + rt * 16;
    const int col  = ct * 16 + l16;

    v8f c = {};
    const float* xrow = H2 + (size_t)(row0 + l16) * HID;
#pragma unroll 4
    for (int kk = 0; kk < HID; kk += 4) {
        const int k = kk + half * 2;
        v2f a = *(const v2f*)(xrow + k);
        v2f b;
        b.x = W[(size_t)k * NCLS + col];
        b.y = W[(size_t)(k + 1) * NCLS + col];
        c = __builtin_amdgcn_wmma_f32_16x16x4_f32(false, a, false, b, (short)0, c, false, false);
    }
#pragma unroll
    for (int v = 0; v < 8; ++v)
        HT[(size_t)(row0 + v + half * 8) * NCLS + col] = c[v];
}

// ---------------- attention logits, layer 2: el2/er2[n] = sum_c h2t[n,c]*a2[c] ----------------
__global__ __launch_bounds__(256) void k_att2(const float* __restrict__ HT,
                                              const float* __restrict__ al,
                                              const float* __restrict__ ar,
                                              float* __restrict__ el, float* __restrict__ er) {
    int t = blockIdx.x * 256 + threadIdx.x;
    if (t >= NN) return;
    const float4* hp = (const float4*)(HT + (size_t)t * NCLS);
    const float4* ap = (const float4*)al;
    const float4* rp = (const float4*)ar;
    float se = 0.f, sr = 0.f;
#pragma unroll
    for (int i = 0; i < NCLS / 4; ++i) {
        const float4 hv = hp[i], av = ap[i], rv = rp[i];
        se += hv.x * av.x + hv.y * av.y + hv.z * av.z + hv.w * av.w;
        sr += hv.x * rv.x + hv.y * rv.y + hv.z * rv.z + hv.w * rv.w;
    }
    el[t] = se; er[t] = sr;
}

// ---------------- layer2 edge pass A ----------------
__global__ __launch_bounds__(256) void k_emax2(const int* __restrict__ src, const int* __restrict__ dst,
                                               const float* __restrict__ el, const float* __restrict__ er,
                                               unsigned* __restrict__ mkey) {
    int t = blockIdx.x * 256 + threadIdx.x;
    if (t >= EE) return;
    const int s = src[t], d = dst[t];
    const float ev = lrelu(el[s] + er[d]);
    atomicMax(&mkey[d], fkey(ev));
}

// ---------------- layer2 edge pass B: 16 threads/edge, 4 dims each ----------------
__global__ __launch_bounds__(256) void k_eagg2(const int* __restrict__ src, const int* __restrict__ dst,
                                               const float* __restrict__ el, const float* __restrict__ er,
                                               const unsigned* __restrict__ mkey,
                                               const float* __restrict__ HT,
                                               float* __restrict__ ssum, float* __restrict__ acc) {
    int t = blockIdx.x * 256 + threadIdx.x;
    if (t >= EE * 16) return;
    const int e = t >> 4, q = t & 15;
    const int s = src[e], d = dst[e];
    const float ev = lrelu(el[s] + er[d]);
    float m = funkey(mkey[d]);
    if (!isfinite(m)) m = 0.f;
    const float p = __expf(ev - m);
    if (q == 0) atomicAdd(&ssum[d], p);
    const float4 hv = ((const float4*)(HT + (size_t)s * NCLS))[q];
    float* ap = acc + (size_t)d * NCLS + q * 4;
    atomicAdd(ap + 0, p * hv.x); atomicAdd(ap + 1, p * hv.y);
    atomicAdd(ap + 2, p * hv.z); atomicAdd(ap + 3, p * hv.w);
}

// ---------------- layer2 finalize: out = acc/s + b (mean over 1 head == identity) ----------------
__global__ __launch_bounds__(256) void k_fin2(const float* __restrict__ acc, const float* __restrict__ ssum,
                                              const float* __restrict__ b, float* __restrict__ out) {
    int t = blockIdx.x * 256 + threadIdx.x;
    if (t >= NN * NCLS) return;
    const int n = t >> 6, cidx = t & 63;
    const float s = ssum[n];
    float v = (s > 0.f) ? acc[t] / s : 0.f;
    out[t] = v + b[cidx];
}

extern "C" void kernel_launch(void* const* d_in, const int* in_sizes, int n_in,
                              void* d_out, int out_size, void* d_ws, size_t ws_size,
                              hipStream_t stream) {
    const float* x   = (const float*)d_in[0];
    const int*   src = (const int*)  d_in[1];
    const int*   dst = (const int*)  d_in[2];
    const float* W1  = (const float*)d_in[3];
    const float* al1 = (const float*)d_in[4];
    const float* ar1 = (const float*)d_in[5];
    const float* b1  = (const float*)d_in[6];
    const float* W2  = (const float*)d_in[7];
    const float* al2 = (const float*)d_in[8];
    const float* ar2 = (const float*)d_in[9];
    const float* b2  = (const float*)d_in[10];
    float* out = (float*)d_out;

    size_t off = 0;
    auto take = [&](size_t nfloats) -> float* {
        float* p = (float*)((char*)d_ws + off);
        off = (off + nfloats * sizeof(float) + 255) & ~(size_t)255;
        return p;
    };
    float*    h1    = take((size_t)NN * HID);   // reused as h2 (layer-1 output) after k_fin1
    float*    el1   = take((size_t)NN * HEADS);
    float*    er1   = take((size_t)NN * HEADS);
    unsigned* mkey1 = (unsigned*)take((size_t)NN * HEADS);
    float*    s1    = take((size_t)NN * HEADS);
    float*    acc1  = take((size_t)NN * HID);   // reused as acc2 (re-zeroed by k_initB)
    float*    h2t   = take((size_t)NN * NCLS);
    float*    el2   = take((size_t)NN);
    float*    er2   = take((size_t)NN);
    unsigned* mkey2 = (unsigned*)take((size_t)NN);
    float*    s2    = take((size_t)NN);
    float*    h2    = h1;
    float*    acc2  = acc1;

    const int T = 256;
    k_initA<<<(NN * HID + T - 1) / T, T, 0, stream>>>(mkey1, s1, acc1, mkey2, s2);
    k_gemm1<<<NN / 16, T, 0, stream>>>(x, W1, h1);
    k_att1 <<<(NN * HEADS + T - 1) / T, T, 0, stream>>>(h1, al1, ar1, el1, er1);
    k_emax1<<<(EE * HEADS + T - 1) / T, T, 0, stream>>>(src, dst, el1, er1, mkey1);
    k_eagg1<<<(EE * HEADS + T - 1) / T, T, 0, stream>>>(src, dst, el1, er1, mkey1, h1, s1, acc1);
    k_fin1 <<<(NN * HID + T - 1) / T, T, 0, stream>>>(acc1, s1, b1, h2);
    k_gemm2<<<NN / 32, T, 0, stream>>>(h2, W2, h2t);
    k_att2 <<<(NN + T - 1) / T, T, 0, stream>>>(h2t, al2, ar2, el2, er2);
    k_initB<<<(NN * NCLS + T - 1) / T, T, 0, stream>>>(acc2);
    k_emax2<<<(EE + T - 1) / T, T, 0, stream>>>(src, dst, el2, er2, mkey2);
    k_eagg2<<<(EE * 16 + T - 1) / T, T, 0, stream>>>(src, dst, el2, er2, mkey2, h2t, s2, acc2);
    k_fin2 <<<(NN * NCLS + T - 1) / T, T, 0, stream>>>(acc2, s2, b2, out);
}